// BiquadCell_1597727834563
// MI455X (gfx1250) — compile-verified
//
#include <hip/hip_runtime.h>
#include <cstdint>

// Problem constants: inputs [T,B,3] f32, carry [B,2] f32, weights [1,5] f32,
// output [T,B,1] f32.
#define T_STEPS 16384
#define BATCH   2048
#define CH      64                    // timesteps per TDM chunk (tile rows)
#define NCHUNK  (T_STEPS / CH)        // 256
#define NBUF    8                     // LDS ring buffers (8 * 8KB = 64KB)
#define DEPTH   6                     // chunks in flight ahead of consumption

typedef __attribute__((ext_vector_type(4))) uint32_t u32x4;
typedef __attribute__((ext_vector_type(8))) uint32_t u32x8;

// gfx1250 has V_TANH_F32 (single trans op) -- use it; fall back to exp2/rcp.
__device__ __forceinline__ float fast_tanhf(float v) {
#if __has_builtin(__builtin_amdgcn_tanhf)
    return __builtin_amdgcn_tanhf(v);
#else
    float e = __builtin_amdgcn_exp2f(v * -2.8853900817779268f);
    return (1.0f - e) * __builtin_amdgcn_rcpf(1.0f + e);
#endif
}

// ---------------------------------------------------------------------------
// Phase 1: a[t,b] = w0*x0 + w1*x1 + (w2+1)*x2  (skip tap folded into w2).
// Fully parallel, fully coalesced float4 traffic: 402MB read + 134MB write at
// HBM rate (~23us). Result lands in d_out; 134MB < 192MB L2, so phase 2's
// reads are mostly L2 hits.
// ---------------------------------------------------------------------------
__global__ __launch_bounds__(256) void biquad_phase1(
    const float* __restrict__ x, const float* __restrict__ w,
    float* __restrict__ a)
{
    const float w0 = w[0], w1 = w[1], w2p = w[2] + 1.0f;
    size_t i = (size_t)blockIdx.x * 256 + threadIdx.x;   // 8,388,608 threads exactly
    const float4* x4 = (const float4*)x;
    float4 v0 = x4[3 * i + 0];
    float4 v1 = x4[3 * i + 1];
    float4 v2 = x4[3 * i + 2];
    float4 r;
    r.x = fmaf(w0, v0.x, fmaf(w1, v0.y, w2p * v0.z));
    r.y = fmaf(w0, v0.w, fmaf(w1, v1.x, w2p * v1.y));
    r.z = fmaf(w0, v1.z, fmaf(w1, v1.w, w2p * v2.x));
    r.w = fmaf(w0, v2.y, fmaf(w1, v2.z, w2p * v2.w));
    ((float4*)a)[i] = r;
}

// ---------------------------------------------------------------------------
// Phase 2: the serial recurrence. One wave32 per block, one channel per lane
// (64 blocks x 32 lanes = 2048 channels). Each chunk of 64 timesteps for the
// wave's 32 channels is a 64x32 f32 tile of the [16384 x 2048] a-tensor --
// staged into an 8-deep LDS ring by the Tensor Data Mover (one
// tensor_load_to_lds per chunk, TENSORcnt-tracked, EXEC-independent), with
// DEPTH=6 chunks (384 steps, 48KB/wave) in flight. Consumption gates on
// s_wait_tensorcnt k where k = chunks issued after the needed one (TDM ops
// from one wave complete in order). Critical path per step stays
// v_fmac_f32 + v_tanh_f32; the w4*c1 term is 2 steps stale -> off-path.
// ---------------------------------------------------------------------------
__global__ __launch_bounds__(32) void biquad_phase2(
    const float* a, const float* __restrict__ carry,
    const float* __restrict__ w, float* out)
{
    __shared__ float sbuf[NBUF * CH * 32];   // 64KB

    const int lane = threadIdx.x;
    const int b    = blockIdx.x * 32 + lane;

    const float w3 = w[3], w4 = w[4];
    float c0 = carry[2 * b + 0];   // o_{t-1}
    float c1 = carry[2 * b + 1];   // o_{t-2}

    // LDS byte offset of this block's buffer (low 32 bits of a generic
    // shared-space pointer are the LDS offset).
    const uint32_t lds_base   = (uint32_t)(size_t)(&sbuf[0]);
    const uint64_t abase_addr = (uint64_t)(size_t)(a + (size_t)blockIdx.x * 32);

    // D# group1 (constant): workgroup_mask=0, data_size=2 (4B), no barrier,
    // no iterate/pad; tensor_dim0=2048, tensor_dim1=16384, tile_dim0=32,
    // tile_dim1=64, tile_dim2=0, tensor_dim0_stride=2048, dim1_stride unused.
    u32x8 g1;
    g1.s0 = 0x00020000u;                    // data_size=4B at bits[17:16]
    g1.s1 = ((uint32_t)BATCH) << 16;        // tensor_dim0[15:0] at [31:16]
    g1.s2 = ((uint32_t)T_STEPS) << 16;      // tensor_dim0[31:16]=0 | tensor_dim1[15:0]
    g1.s3 = 32u << 16;                      // tensor_dim1[31:16]=0 | tile_dim0=32
    g1.s4 = (uint32_t)CH;                   // tile_dim1=64 | tile_dim2=0
    g1.s5 = (uint32_t)BATCH;                // tensor_dim0_stride[31:0]
    g1.s6 = 0u;                             // stride hi | dim1_stride lo (unused)
    g1.s7 = 0u;

    auto issue_chunk = [&](int c) {
        uint64_t gaddr = abase_addr + (uint64_t)c * (CH * BATCH * 4ull);
        uint32_t laddr = lds_base + (uint32_t)(c & (NBUF - 1)) * (CH * 32 * 4u);
        u32x4 g0;
        g0.x = 1u;                                     // count=1 (valid user D#)
        g0.y = laddr;                                  // lds_addr
        g0.z = (uint32_t)gaddr;                        // global_addr[31:0]
        g0.w = ((uint32_t)(gaddr >> 32) & 0x01ffffffu) // global_addr[56:32]
               | (2u << 30);                           // type=2 ("image")
        asm volatile("tensor_load_to_lds %0, %1"
                     :: "s"(g0), "s"(g1)
                     : "memory");
    };

    // Fill the pipeline: DEPTH chunks outstanding (TENSORcnt max 63 >> 6).
    for (int d = 0; d < DEPTH; ++d) issue_chunk(d);

    for (int c = 0; c < NCHUNK; ++c) {
        if (c + DEPTH < NCHUNK) issue_chunk(c + DEPTH);

        // Chunk c complete once TENSORcnt <= (#tensor ops issued after it).
        int rem   = NCHUNK - 1 - c;
        int after = rem < DEPTH ? rem : DEPTH;
        switch (after) {
            case 6:  asm volatile("s_wait_tensorcnt 6" ::: "memory"); break;
            case 5:  asm volatile("s_wait_tensorcnt 5" ::: "memory"); break;
            case 4:  asm volatile("s_wait_tensorcnt 4" ::: "memory"); break;
            case 3:  asm volatile("s_wait_tensorcnt 3" ::: "memory"); break;
            case 2:  asm volatile("s_wait_tensorcnt 2" ::: "memory"); break;
            case 1:  asm volatile("s_wait_tensorcnt 1" ::: "memory"); break;
            default: asm volatile("s_wait_tensorcnt 0" ::: "memory"); break;
        }

        const int    base = (c & (NBUF - 1)) * (CH * 32) + lane;
        float*       dst  = out + (size_t)c * CH * BATCH + b;
#pragma unroll 8
        for (int i = 0; i < CH; ++i) {
            float av  = sbuf[base + i * 32];          // ds_load_b32, conflict-free
            float pre = fmaf(w4, c1, av);             // off critical path (c1 two steps old)
            float v   = fmaf(w3, c0, pre);            // critical: fma
            float o   = fast_tanhf(v);                // critical: v_tanh_f32
            dst[(size_t)i * BATCH] = o;               // 128B coalesced store per wave
            c1 = c0;
            c0 = o;
        }
    }
}

extern "C" void kernel_launch(void* const* d_in, const int* in_sizes, int n_in,
                              void* d_out, int out_size, void* d_ws, size_t ws_size,
                              hipStream_t stream) {
    (void)in_sizes; (void)n_in; (void)out_size; (void)d_ws; (void)ws_size;
    const float* x     = (const float*)d_in[0];   // [T,B,3]
    const float* carry = (const float*)d_in[1];   // [B,2]
    const float* w     = (const float*)d_in[2];   // [1,5]
    float*       out   = (float*)d_out;           // [T,B,1]

    // Phase 1: 33,554,432 (t,b) pairs, 4 per thread.
    biquad_phase1<<<32768, 256, 0, stream>>>(x, w, out);
    // Phase 2: 2048 channels -> 64 blocks of one wave32 each; in-place on out.
    biquad_phase2<<<64, 32, 0, stream>>>(out, carry, w, out);
}